// TQnetOld_16037407883354
// MI455X (gfx1250) — compile-verified
//
#include <hip/hip_runtime.h>
#include <hip/hip_bf16.h>
#include <math.h>

#define N_NODES   100000
#define N_EDGES   3200000
#define HC        32      // HEADS*OUT_CH
#define HE        8       // HEADS*EDGE_EMB
#define NEG_SLOPE 0.2f

typedef float v2f __attribute__((ext_vector_type(2)));
typedef float v8f __attribute__((ext_vector_type(8)));

// ---------------------------------------------------------------- init
__global__ void gat_init(float* __restrict__ out, const float* __restrict__ bias,
                         unsigned* __restrict__ segord, float* __restrict__ denom) {
    int i = blockIdx.x * blockDim.x + threadIdx.x;
    if (i < N_NODES * HC) out[i] = bias[i & (HC - 1)];
    if (i < N_NODES * 2) { segord[i] = 0u; denom[i] = 0.0f; }
}

// ------------------------------------------------- edge_index -> float copy
__global__ void gat_copy_ei(const int* __restrict__ ei, float* __restrict__ eo) {
    int i = blockIdx.x * blockDim.x + threadIdx.x;
    if (i < 2 * N_EDGES) eo[i] = (float)ei[i];
}

// ---------------------------------------------------------------- ew GEMM
__global__ void gat_ew(const float* __restrict__ ea, const float* __restrict__ eaw,
                       float* __restrict__ ew) {
    __shared__ float w[64];
    if (threadIdx.x < 64) w[threadIdx.x] = eaw[threadIdx.x];
    __syncthreads();
    int e = blockIdx.x * blockDim.x + threadIdx.x;
    if (e >= N_EDGES) return;
    const float4* p = (const float4*)(ea + (size_t)e * 8);
    float4 v0 = p[0], v1 = p[1];
    float a[8] = {v0.x, v0.y, v0.z, v0.w, v1.x, v1.y, v1.z, v1.w};
    float o[8];
#pragma unroll
    for (int n = 0; n < 8; ++n) {
        float s = 0.f;
#pragma unroll
        for (int k = 0; k < 8; ++k) s = fmaf(a[k], w[k * 8 + n], s);
        o[n] = s;
    }
    float4* q = (float4*)(ew + (size_t)e * 8);
    q[0] = make_float4(o[0], o[1], o[2], o[3]);
    q[1] = make_float4(o[4], o[5], o[6], o[7]);
}

// ------------------------------------------------- xw = x @ W via f32 WMMA
// One wave computes a 16x32 tile: C0 (cols 0-15) and C1 (cols 16-31),
// looping K=64 in steps of 4 with V_WMMA_F32_16X16X4_F32.
__global__ void gat_xw_wmma(const float* __restrict__ x, const float* __restrict__ w,
                            float* __restrict__ xw) {
    int wave = (blockIdx.x * blockDim.x + threadIdx.x) >> 5;
    int lane = threadIdx.x & 31;
    const int nwaves = (N_NODES + 15) / 16;
    if (wave >= nwaves) return;          // wave-uniform: EXEC stays all-1s below
    int row0 = wave * 16;
    int m  = lane & 15;
    int kh = lane >> 4;                  // K-half selector (ISA A layout)
    int n  = lane & 15;
    int row = row0 + m;
    if (row >= N_NODES) row = N_NODES - 1;   // clamp loads, keep EXEC full
    const float* xr = x + (size_t)row * 64;

    v8f c0 = {}; v8f c1 = {};
#pragma unroll
    for (int k0 = 0; k0 < 64; k0 += 4) {
        int k = k0 + 2 * kh;
        v2f a;  a[0] = xr[k];            a[1] = xr[k + 1];
        v2f b0; b0[0] = w[k * 32 + n];       b0[1] = w[(k + 1) * 32 + n];
        v2f b1; b1[0] = w[k * 32 + 16 + n];  b1[1] = w[(k + 1) * 32 + 16 + n];
        c0 = __builtin_amdgcn_wmma_f32_16x16x4_f32(false, a, false, b0,
                                                   (short)0, c0, false, false);
        c1 = __builtin_amdgcn_wmma_f32_16x16x4_f32(false, a, false, b1,
                                                   (short)0, c1, false, false);
    }
    int mh = lane >> 4;                  // C layout: lanes 16-31 hold M=8..15
#pragma unroll
    for (int r = 0; r < 8; ++r) {
        int mm = row0 + r + 8 * mh;
        if (mm < N_NODES) {
            xw[(size_t)mm * 32 + n]      = c0[r];
            xw[(size_t)mm * 32 + 16 + n] = c1[r];
        }
    }
}

// --------------------------------------------- ordered-uint float max helpers
__device__ __forceinline__ unsigned f2ord(float f) {
    unsigned u = __float_as_uint(f);
    return (u & 0x80000000u) ? ~u : (u | 0x80000000u);
}
__device__ __forceinline__ float ord2f(unsigned o) {
    unsigned u = (o & 0x80000000u) ? (o & 0x7fffffffu) : ~o;
    return __uint_as_float(u);
}

// ---------------------------------------- pass A: alpha + segment-max (atomic)
__global__ void gat_alpha(const float* __restrict__ xw, const float* __restrict__ ew,
                          const int* __restrict__ ei, const float* __restrict__ att,
                          float* __restrict__ alpha, unsigned* __restrict__ segord) {
    __shared__ float s_att[72];
    if (threadIdx.x < 72) s_att[threadIdx.x] = att[threadIdx.x];
    __syncthreads();
    int e = blockIdx.x * blockDim.x + threadIdx.x;
    if (e >= N_EDGES) return;
    int src = ei[e];
    int dst = ei[N_EDGES + e];
    const float* xi = xw + (size_t)dst * 32;
    const float* xj = xw + (size_t)src * 32;
    const float* ee = ew + (size_t)e * 8;
#pragma unroll
    for (int h = 0; h < 2; ++h) {
        const float* at = s_att + h * 36;
        float s = 0.f;
#pragma unroll
        for (int c = 0; c < 16; ++c)
            s = fmaf(xi[h * 16 + c], at[c], fmaf(xj[h * 16 + c], at[16 + c], s));
#pragma unroll
        for (int c = 0; c < 4; ++c)
            s = fmaf(ee[h * 4 + c], at[32 + c], s);
        s = (s >= 0.f) ? s : NEG_SLOPE * s;           // leaky relu
        alpha[(size_t)e * 2 + h] = s;
        atomicMax(&segord[dst * 2 + h], f2ord(s));
    }
}

// ------------------------------------- pass B: exp(alpha - max), denom (atomic)
__global__ void gat_expsum(const int* __restrict__ ei, const unsigned* __restrict__ segord,
                           float* __restrict__ alpha, float* __restrict__ denom) {
    int e = blockIdx.x * blockDim.x + threadIdx.x;
    if (e >= N_EDGES) return;
    int dst = ei[N_EDGES + e];
#pragma unroll
    for (int h = 0; h < 2; ++h) {
        float mx = ord2f(segord[dst * 2 + h]);
        float ex = __expf(alpha[(size_t)e * 2 + h] - mx);
        alpha[(size_t)e * 2 + h] = ex;
        atomicAdd(&denom[dst * 2 + h], ex);
    }
}

// ------------------------------------------- pass C: normalized scatter-add
__global__ void gat_scatter(const int* __restrict__ ei, const float* __restrict__ xw,
                            const float* __restrict__ alpha, const float* __restrict__ denom,
                            float* __restrict__ out) {
    int e = blockIdx.x * blockDim.x + threadIdx.x;
    if (e >= N_EDGES) return;
    int src = ei[e];
    int dst = ei[N_EDGES + e];
    const float* xj = xw + (size_t)src * 32;
    float* od = out + (size_t)dst * 32;
#pragma unroll
    for (int h = 0; h < 2; ++h) {
        float wgt = alpha[(size_t)e * 2 + h] / (denom[dst * 2 + h] + 1e-16f);
#pragma unroll
        for (int c = 0; c < 16; ++c)
            atomicAdd(&od[h * 16 + c], xj[h * 16 + c] * wgt);
    }
}

// ---------------------------------------------------------------- launch
extern "C" void kernel_launch(void* const* d_in, const int* in_sizes, int n_in,
                              void* d_out, int out_size, void* d_ws, size_t ws_size,
                              hipStream_t stream) {
    const float* x    = (const float*)d_in[0];
    const int*   ei   = (const int*)d_in[1];
    const float* ea   = (const float*)d_in[2];
    const float* w    = (const float*)d_in[3];
    const float* eaw  = (const float*)d_in[4];
    const float* att  = (const float*)d_in[5];
    const float* bias = (const float*)d_in[6];

    // d_out layout: [out | edge_index as float | ew]
    float* out    = (float*)d_out;
    float* ei_out = out + (size_t)N_NODES * HC;
    float* ew_out = ei_out + (size_t)2 * N_EDGES;

    // workspace layout
    float*    xw     = (float*)d_ws;                               // N_NODES*32
    unsigned* segord = (unsigned*)(xw + (size_t)N_NODES * HC);     // N_NODES*2
    float*    denom  = (float*)(segord + (size_t)N_NODES * 2);     // N_NODES*2
    float*    alpha  = denom + (size_t)N_NODES * 2;                // N_EDGES*2

    const int B = 256;
    int g_init = (N_NODES * HC + B - 1) / B;
    int g_ei   = (2 * N_EDGES + B - 1) / B;
    int g_edge = (N_EDGES + B - 1) / B;
    int nwaves = (N_NODES + 15) / 16;
    int g_xw   = (nwaves * 32 + B - 1) / B;

    gat_init   <<<g_init, B, 0, stream>>>(out, bias, segord, denom);
    gat_copy_ei<<<g_ei,   B, 0, stream>>>(ei, ei_out);
    gat_ew     <<<g_edge, B, 0, stream>>>(ea, eaw, ew_out);
    gat_xw_wmma<<<g_xw,   B, 0, stream>>>(x, w, xw);
    gat_alpha  <<<g_edge, B, 0, stream>>>(xw, ew_out, ei, att, alpha, segord);
    gat_expsum <<<g_edge, B, 0, stream>>>(ei, segord, alpha, denom);
    gat_scatter<<<g_edge, B, 0, stream>>>(ei, xw, alpha, denom, out);
}